// WarpLayer_34840774705571
// MI455X (gfx1250) — compile-verified
//
#include <hip/hip_runtime.h>

// Backward bilinear warp, NHWC, B=8 H=384 W=512 C=32, fp32.
// Memory-bound gather: ~415 MB traffic -> ~18us roofline at 23.3 TB/s.
// Layout: 8 lanes per pixel, each lane owns a float4 channel chunk.
//   - 4 corner gathers per lane are global_load_b128, 128B contiguous/pixel.
//   - Output store is nontemporal b128 (write-once; keep L2 for the image).
//   - Flow load is nontemporal b64 (read-once) + global_prefetch_b8 ahead.

typedef float v4f __attribute__((ext_vector_type(4)));
typedef float v2f __attribute__((ext_vector_type(2)));

__global__ __launch_bounds__(256) void warp_bilinear_kernel(
    const float* __restrict__ image,
    const float* __restrict__ flow,
    float* __restrict__ out) {
  constexpr int H = 384, W = 512, HW = H * W;

  const int tid   = threadIdx.x;
  const int pix   = blockIdx.x * 32 + (tid >> 3);  // 32 pixels per block
  const int chunk = tid & 7;                        // float4 chunk of C=32

  // Prefetch the sequential flow stream ~4096 pixels ahead (speculative:
  // past-end addresses are silently dropped). Lowers to global_prefetch_b8.
  __builtin_prefetch((const char*)(flow + 2 * (pix + 4096)), 0, 0);

  // Per-pixel flow (all 8 lanes of a pixel read the same 8B; L1 broadcasts).
  // Read-once stream -> nontemporal.
  const v2f f = __builtin_nontemporal_load((const v2f*)flow + pix);
  const float fy = f.x;  // flow[...,0] -> subtracted from grid_y
  const float fx = f.y;  // flow[...,1] -> subtracted from grid_x

  const int b   = pix / HW;
  const int rem = pix - b * HW;
  const int y   = rem >> 9;        // W = 512
  const int x   = rem & (W - 1);

  const float qy = (float)y - fy;
  const float qx = (float)x - fx;

  float fly = fminf(fmaxf(floorf(qy), 0.0f), (float)(H - 2));
  float flx = fminf(fmaxf(floorf(qx), 0.0f), (float)(W - 2));
  const float ay = fminf(fmaxf(qy - fly, 0.0f), 1.0f);
  const float ax = fminf(fmaxf(qx - flx, 0.0f), 1.0f);
  const int y0 = (int)fly;
  const int x0 = (int)flx;

  // Gather 4 corners: base address in float4 units; byte offsets < 2^31.
  const v4f* __restrict__ img4 = (const v4f*)image;
  const int base = (b * HW + y0 * W + x0) * 8 + chunk;  // C=32 -> 8 float4s
  const v4f tl = img4[base];
  const v4f tr = img4[base + 8];
  const v4f bl = img4[base + 8 * W];
  const v4f br = img4[base + 8 * W + 8];

  const v4f top = tl + ax * (tr - tl);
  const v4f bot = bl + ax * (br - bl);
  const v4f res = top + ay * (bot - top);

  // Write-once output: nontemporal b128 store, 512B contiguous per wave.
  __builtin_nontemporal_store(res, (v4f*)out + (pix * 8 + chunk));
}

extern "C" void kernel_launch(void* const* d_in, const int* in_sizes, int n_in,
                              void* d_out, int out_size, void* d_ws, size_t ws_size,
                              hipStream_t stream) {
  const float* image = (const float*)d_in[0];  // [8,384,512,32] fp32
  const float* flow  = (const float*)d_in[1];  // [8,384,512,2]  fp32
  float* out = (float*)d_out;                  // [8,384,512,32] fp32

  constexpr int P = 8 * 384 * 512;             // 1,572,864 pixels
  const dim3 grid(P / 32);                     // 32 pixels per 256-thread block
  const dim3 block(256);                       // 8 waves (wave32) per block
  hipLaunchKernelGGL(warp_bilinear_kernel, grid, block, 0, stream,
                     image, flow, out);
}